// MoleculeGNN_28965259444469
// MI455X (gfx1250) — compile-verified
//
#include <hip/hip_runtime.h>
#include <hip/hip_bf16.h>
#include <math.h>

// ---------------------------------------------------------------------------
// MoleculeGNN (3-layer GINE, H=128) for gfx1250 (MI455X, wave32, WMMA).
//  - edge message/scatter: wave-per-edge, float4 gathers, f32 global atomics
//  - node MLP: fused dual 128x128 GEMM on V_WMMA_F32_16X16X4_F32, LDS staging
//  - encoder / pooling / head: VALU (negligible cost, memory-bound pipeline)
// ---------------------------------------------------------------------------

#define N_NODES  200000
#define N_EDGES  800000
#define N_GRAPHS 8192
#define HD       128
// 1/sqrt(1 + 1e-5)
#define INVSTD   0.9999950000374997f

typedef float v2f __attribute__((ext_vector_type(2)));
typedef float v8f __attribute__((ext_vector_type(8)));

// ---------------------------------------------------------------------------
// Encoder: h = x @ W_enc + b_enc    (x: [N,78], W: [78,128])
// Block = 256 threads, 16 nodes per block; x tile staged in LDS.
// ---------------------------------------------------------------------------
__global__ __launch_bounds__(256) void enc_kernel(
    const float* __restrict__ x, const float* __restrict__ W,
    const float* __restrict__ b, float* __restrict__ h)
{
    __shared__ float xs[16 * 78];
    const int t  = threadIdx.x;
    const int n0 = blockIdx.x * 16;
    for (int i = t; i < 16 * 78; i += 256) {
        int r = i / 78, c = i - r * 78;
        xs[i] = x[(size_t)(n0 + r) * 78 + c];
    }
    __syncthreads();
#pragma unroll
    for (int i = 0; i < 8; ++i) {
        int o   = t + i * 256;       // 0..2047
        int row = o >> 7;
        int col = o & 127;
        float a = b[col];
        const float* xr = &xs[row * 78];
        for (int k = 0; k < 78; ++k)
            a = fmaf(xr[k], W[k * HD + col], a);
        h[(size_t)(n0 + row) * HD + col] = a;
    }
}

// ---------------------------------------------------------------------------
// Edge message + scatter:
//   m = relu(h[src] + edge_attr @ We_l + be_l);  agg[dst] += m  (atomics)
// One wave per edge (grid-strided). We_l (10x128) + be_l staged in LDS.
// ---------------------------------------------------------------------------
__global__ __launch_bounds__(256) void msg_kernel(
    const int* __restrict__ src, const int* __restrict__ dst,
    const float* __restrict__ ea, const float* __restrict__ h,
    const float* __restrict__ We, const float* __restrict__ be,
    float* __restrict__ agg)
{
    __shared__ float Ws[10 * HD];
    __shared__ float bs[HD];
    const int t = threadIdx.x;
    for (int i = t; i < 10 * HD; i += 256) Ws[i] = We[i];
    if (t < HD) bs[t] = be[t];
    __syncthreads();

    const int lane = t & 31;
    const int gw   = blockIdx.x * 8 + (t >> 5);
    const int nw   = gridDim.x * 8;
    const int c0   = lane * 4;

    for (int e = gw; e < N_EDGES; e += nw) {
        const int s = src[e];
        const int d = dst[e];
        float a[10];
#pragma unroll
        for (int k = 0; k < 10; ++k) a[k] = ea[(size_t)e * 10 + k];

        const float4 hv = *(const float4*)&h[(size_t)s * HD + c0];
        float4 m        = *(const float4*)&bs[c0];
        m.x += hv.x; m.y += hv.y; m.z += hv.z; m.w += hv.w;
#pragma unroll
        for (int k = 0; k < 10; ++k) {
            const float4 w = *(const float4*)&Ws[k * HD + c0];
            m.x = fmaf(a[k], w.x, m.x);
            m.y = fmaf(a[k], w.y, m.y);
            m.z = fmaf(a[k], w.z, m.z);
            m.w = fmaf(a[k], w.w, m.w);
        }
        float* o = &agg[(size_t)d * HD + c0];
        atomicAdd(o + 0, fmaxf(m.x, 0.0f));
        atomicAdd(o + 1, fmaxf(m.y, 0.0f));
        atomicAdd(o + 2, fmaxf(m.z, 0.0f));
        atomicAdd(o + 3, fmaxf(m.w, 0.0f));
    }
}

// ---------------------------------------------------------------------------
// Fused GINE node MLP + BatchNorm(eval) + relu, on fp32 WMMA:
//   t = relu(z @ W1 + b1);  y = (t @ W2 + b2) * (invstd*gamma) + beta;
//   h = relu(y)
// Block = 256 threads (8 waves), 64 nodes per block.
// Wave w owns N-tile [16w,16w+16) and loops 4 M-tiles (B frag reused 4x).
// ---------------------------------------------------------------------------
__global__ __launch_bounds__(256) void mlp_kernel(
    const float* __restrict__ agg, float* __restrict__ hout,
    const float* __restrict__ W1, const float* __restrict__ b1,
    const float* __restrict__ W2, const float* __restrict__ b2,
    const float* __restrict__ gamma, const float* __restrict__ beta)
{
    __shared__ float zA[64 * HD];   // input tile  (32 KB)
    __shared__ float zT[64 * HD];   // intermediate (32 KB)

    const int t    = threadIdx.x;
    const int lane = t & 31;
    const int wv   = t >> 5;        // 0..7 -> N tile
    const int lr   = lane & 15;     // row (A) / col (B,C)
    const int lh   = lane >> 4;     // K half select
    const int m0   = blockIdx.x * 64;

    // cooperative load of the 64x128 input tile
    {
        const float4* g = (const float4*)(agg + (size_t)m0 * HD);
        float4* l = (float4*)zA;
#pragma unroll
        for (int i = 0; i < 8; ++i) l[t + i * 256] = g[t + i * 256];
    }
    __syncthreads();

    const int ncol = wv * 16 + lr;

    // ---------------- GEMM 1: t = relu(zA @ W1 + b1) ----------------
    v8f acc[4];
    {
        const float bias = b1[ncol];
#pragma unroll
        for (int mt = 0; mt < 4; ++mt)
#pragma unroll
            for (int j = 0; j < 8; ++j) acc[mt][j] = bias;
    }
#pragma unroll 4
    for (int kb = 0; kb < HD; kb += 4) {
        const int krow = kb + 2 * lh;
        v2f bf;
        bf.x = W1[krow * HD + ncol];
        bf.y = W1[(krow + 1) * HD + ncol];
#pragma unroll
        for (int mt = 0; mt < 4; ++mt) {
            const v2f af = *(const v2f*)&zA[(mt * 16 + lr) * HD + krow];
            acc[mt] = __builtin_amdgcn_wmma_f32_16x16x4_f32(
                false, af, false, bf, (short)0, acc[mt], false, false);
        }
    }
#pragma unroll
    for (int mt = 0; mt < 4; ++mt)
#pragma unroll
        for (int j = 0; j < 8; ++j) {
            const int row = mt * 16 + j + 8 * lh;
            zT[row * HD + ncol] = fmaxf(acc[mt][j], 0.0f);
        }
    __syncthreads();

    // ---------------- GEMM 2 + BN + relu ----------------
    {
        const float bias = b2[ncol];
#pragma unroll
        for (int mt = 0; mt < 4; ++mt)
#pragma unroll
            for (int j = 0; j < 8; ++j) acc[mt][j] = bias;
    }
#pragma unroll 4
    for (int kb = 0; kb < HD; kb += 4) {
        const int krow = kb + 2 * lh;
        v2f bf;
        bf.x = W2[krow * HD + ncol];
        bf.y = W2[(krow + 1) * HD + ncol];
#pragma unroll
        for (int mt = 0; mt < 4; ++mt) {
            const v2f af = *(const v2f*)&zT[(mt * 16 + lr) * HD + krow];
            acc[mt] = __builtin_amdgcn_wmma_f32_16x16x4_f32(
                false, af, false, bf, (short)0, acc[mt], false, false);
        }
    }
    const float sc = INVSTD * gamma[ncol];
    const float sh = beta[ncol];
#pragma unroll
    for (int mt = 0; mt < 4; ++mt)
#pragma unroll
        for (int j = 0; j < 8; ++j) {
            const int row = mt * 16 + j + 8 * lh;
            hout[(size_t)(m0 + row) * HD + ncol] =
                fmaxf(fmaf(acc[mt][j], sc, sh), 0.0f);
        }
}

// ---------------------------------------------------------------------------
// Global mean-pool, pass 1: sums[batch[n]] += h[n];  cnt[batch[n]] += 1
// ---------------------------------------------------------------------------
__global__ __launch_bounds__(256) void pool_kernel(
    const float* __restrict__ h, const int* __restrict__ batch,
    float* __restrict__ sums, float* __restrict__ cnt)
{
    const int lane = threadIdx.x & 31;
    const int gw   = (blockIdx.x * 256 + threadIdx.x) >> 5;
    const int nw   = (gridDim.x * 256) >> 5;
    const int c0   = lane * 4;
    for (int n = gw; n < N_NODES; n += nw) {
        const int b = batch[n];
        const float4 hv = *(const float4*)&h[(size_t)n * HD + c0];
        float* s = &sums[(size_t)b * HD + c0];
        atomicAdd(s + 0, hv.x);
        atomicAdd(s + 1, hv.y);
        atomicAdd(s + 2, hv.z);
        atomicAdd(s + 3, hv.w);
        if (lane == 0) atomicAdd(&cnt[b], 1.0f);
    }
}

// ---------------------------------------------------------------------------
// Head: out = sigmoid(relu(mean @ Wh1 + bh1) @ Wh2 + bh2)
// One wave per graph; each lane owns 2 of the 64 hidden units.
// ---------------------------------------------------------------------------
__global__ __launch_bounds__(256) void head_kernel(
    const float* __restrict__ sums, const float* __restrict__ cnt,
    const float* __restrict__ Wh1, const float* __restrict__ bh1,
    const float* __restrict__ Wh2, const float* __restrict__ bh2,
    float* __restrict__ out)
{
    const int lane = threadIdx.x & 31;
    const int g    = blockIdx.x * 8 + (threadIdx.x >> 5);
    if (g >= N_GRAPHS) return;

    const float inv = 1.0f / fmaxf(cnt[g], 1.0f);
    float h0 = bh1[lane];
    float h1 = bh1[lane + 32];
    const float* p = &sums[(size_t)g * HD];
    for (int k = 0; k < HD; ++k) {
        const float pk = p[k] * inv;
        h0 = fmaf(pk, Wh1[k * 64 + lane], h0);
        h1 = fmaf(pk, Wh1[k * 64 + lane + 32], h1);
    }
    h0 = fmaxf(h0, 0.0f);
    h1 = fmaxf(h1, 0.0f);
    float part = h0 * Wh2[lane] + h1 * Wh2[lane + 32];
#pragma unroll
    for (int off = 16; off > 0; off >>= 1)
        part += __shfl_xor(part, off, 32);
    if (lane == 0)
        out[g] = 1.0f / (1.0f + __expf(-(part + bh2[0])));
}

// ---------------------------------------------------------------------------
extern "C" void kernel_launch(void* const* d_in, const int* in_sizes, int n_in,
                              void* d_out, int out_size, void* d_ws, size_t ws_size,
                              hipStream_t stream) {
    (void)in_sizes; (void)n_in; (void)out_size; (void)ws_size;

    const float* x     = (const float*)d_in[0];
    const int*   eidx  = (const int*)  d_in[1];
    const float* ea    = (const float*)d_in[2];
    const int*   batch = (const int*)  d_in[3];
    const float* W_enc = (const float*)d_in[4];
    const float* b_enc = (const float*)d_in[5];
    const float* We    = (const float*)d_in[6];
    const float* be    = (const float*)d_in[7];
    const float* W1    = (const float*)d_in[8];
    const float* b1    = (const float*)d_in[9];
    const float* W2    = (const float*)d_in[10];
    const float* b2    = (const float*)d_in[11];
    const float* gamma = (const float*)d_in[12];
    const float* beta  = (const float*)d_in[13];
    const float* Wh1   = (const float*)d_in[14];
    const float* bh1   = (const float*)d_in[15];
    const float* Wh2   = (const float*)d_in[16];
    const float* bh2   = (const float*)d_in[17];
    float* out = (float*)d_out;

    const size_t NH_BYTES = (size_t)N_NODES * HD * sizeof(float); // 102.4 MB
    char* ws = (char*)d_ws;
    float* h    = (float*)ws;
    float* agg  = (float*)(ws + NH_BYTES);
    float* sums = (float*)(ws + 2 * NH_BYTES);               // [G, H]
    float* cnt  = (float*)(ws + 2 * NH_BYTES +
                           (size_t)N_GRAPHS * HD * sizeof(float)); // [G]

    const int* src = eidx;
    const int* dst = eidx + N_EDGES;

    enc_kernel<<<N_NODES / 16, 256, 0, stream>>>(x, W_enc, b_enc, h);

    for (int l = 0; l < 3; ++l) {
        hipMemcpyAsync(agg, h, NH_BYTES, hipMemcpyDeviceToDevice, stream);
        msg_kernel<<<4096, 256, 0, stream>>>(src, dst, ea, h,
                                             We + (size_t)l * 10 * HD,
                                             be + (size_t)l * HD, agg);
        mlp_kernel<<<N_NODES / 64, 256, 0, stream>>>(agg, h,
                                                     W1 + (size_t)l * HD * HD,
                                                     b1 + (size_t)l * HD,
                                                     W2 + (size_t)l * HD * HD,
                                                     b2 + (size_t)l * HD,
                                                     gamma + (size_t)l * HD,
                                                     beta + (size_t)l * HD);
    }

    hipMemsetAsync(sums, 0, ((size_t)N_GRAPHS * HD + N_GRAPHS) * sizeof(float),
                   stream);
    pool_kernel<<<2048, 256, 0, stream>>>(h, batch, sums, cnt);
    head_kernel<<<N_GRAPHS / 8, 256, 0, stream>>>(sums, cnt, Wh1, bh1, Wh2, bh2,
                                                  out);
}